// LocalConv_11510512353617
// MI455X (gfx1250) — compile-verified
//
#include <hip/hip_runtime.h>

typedef float v2f __attribute__((ext_vector_type(2)));
typedef float v8f __attribute__((ext_vector_type(8)));

#define NC 96
#define KS 7
#define NB 32
#define NH 112
#define NW 112
#define HWSZ (NH*NW)
#define TILE 16
#define HALO 22            // TILE + 2*3
#define CHUNK 16
#define NCHUNK 6
#define XROW 23            // halo row stride (floats)
#define XCH 513            // per-channel halo stride (odd -> bank-friendly)
#define FST 18             // feat pos stride (even -> 8B-aligned k-pairs)
#define HKST 52            // 7x7 kernel stride
#define YST 257            // staging o stride

#define OFF_X0 0                         // 8208 floats
#define OFF_X1 (OFF_X0 + CHUNK*XCH)      // 8208
#define OFF_F  (OFF_X1 + CHUNK*XCH)      // 16416
#define OFF_W  (OFF_F + 256*FST)         // 21024
#define OFF_HK (OFF_W + (CHUNK/2)*NC*2)  // 22560
#define OFF_B  (OFF_HK + CHUNK*HKST)     // 23392
#define OFF_S  (OFF_B + NC)              // 23488
#define SMEM_FLOATS (OFF_S + 2*NC)       // 23680 floats = 94,720 B (<320KB WGP LDS)

#if __has_builtin(__builtin_amdgcn_global_load_async_to_lds_b32) && \
    __has_builtin(__builtin_amdgcn_s_wait_asynccnt)
#define HAVE_ASYNC 1
typedef __attribute__((address_space(1))) int* gas_i32;
typedef __attribute__((address_space(3))) int* lds_i32;
#else
#define HAVE_ASYNC 0
#endif

__global__ void zero_stats_k(float* stats) {
  int t = threadIdx.x;
  if (t < 2*NC) stats[t] = 0.0f;
}

// Stage one 16-channel halo tile into LDS. Async path: in-bounds lanes issue
// global_load_async_to_lds_b32 (ASYNCcnt), OOB lanes ds_store zero directly.
__device__ __forceinline__ void stage_x_chunk(const float* __restrict__ x,
                                              float* __restrict__ smbase,
                                              int b, int c0, int th0, int tw0,
                                              int tid) {
  for (int idx = tid; idx < CHUNK*HALO*HALO; idx += 256) {
    int cc  = idx / (HALO*HALO);
    int rem = idx - cc*(HALO*HALO);
    int r   = rem / HALO;
    int q   = rem - r*HALO;
    int hh = th0 + r - 3, ww = tw0 + q - 3;
    float* lp = smbase + cc*XCH + r*XROW + q;
    bool inb = ((unsigned)hh < (unsigned)NH) && ((unsigned)ww < (unsigned)NW);
#if HAVE_ASYNC
    if (inb) {
      const float* gp = x + (((size_t)(b*NC + c0 + cc))*NH + hh)*NW + ww;
      __builtin_amdgcn_global_load_async_to_lds_b32((gas_i32)gp, (lds_i32)lp,
                                                    /*offset=*/0, /*cpol=*/0);
    } else {
      *lp = 0.f;
    }
#else
    float v = 0.f;
    if (inb) v = x[(((size_t)(b*NC + c0 + cc))*NH + hh)*NW + ww];
    *lp = v;
#endif
  }
}

__global__ __launch_bounds__(256) void fused_conv_mix_k(
    const float* __restrict__ hk, const float* __restrict__ x,
    const float* __restrict__ wmix, const float* __restrict__ bmix,
    float* __restrict__ y, float* __restrict__ stats)
{
  __shared__ __align__(16) float sm[SMEM_FLOATS];
  const int tid = threadIdx.x;
  const int blk = blockIdx.x;
  const int b   = blk / 49;
  const int tt  = blk % 49;
  const int th0 = (tt / 7) * TILE;
  const int tw0 = (tt % 7) * TILE;

  const int lane = tid & 31;   // wave32
  const int wv   = tid >> 5;   // 0..7
  const int lm   = lane & 15;
  const int lh   = lane >> 4;

  v8f acc[2][6];
  const v8f vzero = {0.f,0.f,0.f,0.f,0.f,0.f,0.f,0.f};
#pragma unroll
  for (int a = 0; a < 2; ++a)
#pragma unroll
    for (int o = 0; o < 6; ++o) acc[a][o] = vzero;

  if (tid < 2*NC) sm[OFF_S + tid] = 0.f;
  if (tid < NC)   sm[OFF_B + tid] = bmix[tid];

  // preload chunk 0 into buffer 0 (async if available)
  stage_x_chunk(x, &sm[OFF_X0], b, 0, th0, tw0, tid);

#pragma unroll 1
  for (int ch = 0; ch < NCHUNK; ++ch) {
    const int c0  = ch * CHUNK;
    const int cur = ch & 1;
    float* xcur = cur ? &sm[OFF_X1] : &sm[OFF_X0];
    float* xnxt = cur ? &sm[OFF_X0] : &sm[OFF_X1];
#if HAVE_ASYNC
    __builtin_amdgcn_s_wait_asynccnt(0);   // chunk ch's halo tile landed in LDS
#endif
    __syncthreads();   // B1: x(ch) visible to all waves; chunk ch-1 compute done

    // overlap: issue chunk ch+1's halo loads while we compute chunk ch
    if (ch + 1 < NCHUNK)
      stage_x_chunk(x, xnxt, b, c0 + CHUNK, th0, tw0, tid);

    // ---- stage depthwise weights ----
    for (int idx = tid; idx < CHUNK*49; idx += 256) {
      int cc = idx / 49, k = idx - cc*49;
      sm[OFF_HK + cc*HKST + k] = hk[(c0 + cc)*49 + k];
    }
    // ---- stage w_mix^T, k-pair interleaved: wT2[kpair][o] = {w[o][2kp], w[o][2kp+1]} ----
    for (int idx = tid; idx < CHUNK*NC; idx += 256) {
      int cc = idx / NC, o = idx - cc*NC;
      sm[OFF_W + ((cc >> 1)*NC + o)*2 + (cc & 1)] = wmix[o*NC + c0 + cc];
    }
    __syncthreads();   // B2: hk / wT2 ready

    // ---- depthwise 7x7 conv: one thread = one spatial position ----
    {
      const int pr = tid >> 4, pc = tid & 15;
#pragma unroll 1
      for (int cc = 0; cc < CHUNK; ++cc) {
        float a0 = 0.f;
#pragma unroll
        for (int i = 0; i < KS; ++i)
#pragma unroll
          for (int j = 0; j < KS; ++j)
            a0 += xcur[cc*XCH + (pr+i)*XROW + (pc+j)] *
                  sm[OFF_HK + cc*HKST + i*KS + j];
        sm[OFF_F + tid*FST + cc] = a0;
      }
    }
    __syncthreads();   // B3: feat ready

    // ---- 1x1 mixer GEMM via f32 WMMA: D[pos,o] += feat[pos,c] * w[o,c] ----
#pragma unroll
    for (int ks = 0; ks < CHUNK/4; ++ks) {
      v2f Amat[2];
#pragma unroll
      for (int a = 0; a < 2; ++a) {
        int pos = (2*wv + a)*16 + lm;            // A: M = lane&15, kpair = lane>>4
        Amat[a] = *(const v2f*)&sm[OFF_F + pos*FST + ks*4 + 2*lh];
      }
#pragma unroll
      for (int ot = 0; ot < 6; ++ot) {
        v2f Bmat = *(const v2f*)&sm[OFF_W + ((ks*2 + lh)*NC + ot*16 + lm)*2];
#pragma unroll
        for (int a = 0; a < 2; ++a)
          acc[a][ot] = __builtin_amdgcn_wmma_f32_16x16x4_f32(
              false, Amat[a], false, Bmat, (short)0, acc[a][ot], false, false);
      }
    }
  }

  // ---- epilogue: bias, LDS transpose (overlays X0), coalesced store, stat reduce ----
#pragma unroll 1
  for (int ot = 0; ot < 6; ++ot) {
    __syncthreads();
    float bias = sm[OFF_B + ot*16 + lm];
#pragma unroll
    for (int a = 0; a < 2; ++a) {
      int posbase = (2*wv + a)*16 + 8*lh;        // C: VGPR r -> M = r + 8*(lane>>4)
#pragma unroll
      for (int r = 0; r < 8; ++r)
        sm[OFF_X0 + lm*YST + posbase + r] = acc[a][ot][r] + bias;
    }
    __syncthreads();
    int oo = tid >> 4;
    int o  = ot*16 + oo;
    int pc = tid & 15;
    float s = 0.f, s2 = 0.f;
    const size_t obase = ((size_t)(b*NC + o))*HWSZ;
#pragma unroll
    for (int i = 0; i < 16; ++i) {
      float v = sm[OFF_X0 + oo*YST + pc + 16*i];  // pos = i*16 + pc
      s += v; s2 += v*v;
      y[obase + (size_t)(th0 + i)*NW + (tw0 + pc)] = v;
    }
    atomicAdd(&sm[OFF_S + o], s);
    atomicAdd(&sm[OFF_S + NC + o], s2);
  }
  __syncthreads();
  if (tid < 2*NC) atomicAdd(&stats[tid], sm[OFF_S + tid]);
}

__global__ void finalize_stats_k(float* stats) {
  int t = threadIdx.x;
  if (t < NC) {
    const float invN = 1.0f / (float)((size_t)NB * HWSZ);
    float m = stats[t] * invN;
    float v = stats[NC + t] * invN - m*m;
    stats[2*NC + t] = m;
    stats[3*NC + t] = rsqrtf(v + 1e-5f);
  }
}

__device__ __forceinline__ float gelu_exact(float x) {
  return 0.5f * x * (1.0f + erff(x * 0.70710678118654752440f));
}

__global__ __launch_bounds__(256) void bn_gelu_k(float* __restrict__ y,
                                                 const float* __restrict__ stats,
                                                 long long total4) {
  long long i = (long long)blockIdx.x * blockDim.x + threadIdx.x;
  if (i >= total4) return;
  long long e = i * 4;
  int o = (int)((e / HWSZ) % NC);      // HW divisible by 4 -> o uniform in float4
  float m = stats[2*NC + o];
  float r = stats[3*NC + o];
  float4 v = ((const float4*)y)[i];
  v.x = gelu_exact((v.x - m) * r);
  v.y = gelu_exact((v.y - m) * r);
  v.z = gelu_exact((v.z - m) * r);
  v.w = gelu_exact((v.w - m) * r);
  ((float4*)y)[i] = v;
}

extern "C" void kernel_launch(void* const* d_in, const int* in_sizes, int n_in,
                              void* d_out, int out_size, void* d_ws, size_t ws_size,
                              hipStream_t stream) {
  (void)in_sizes; (void)n_in; (void)out_size; (void)ws_size;
  const float* hk   = (const float*)d_in[0];   // [96,1,7,7]
  const float* x    = (const float*)d_in[1];   // [32,96,112,112]
  const float* wmix = (const float*)d_in[2];   // [96,96]
  const float* bmix = (const float*)d_in[3];   // [96]
  float* y     = (float*)d_out;                // [32,96,112,112]
  float* stats = (float*)d_ws;                 // [sum96|sumsq96|mean96|rstd96]

  zero_stats_k<<<dim3(1), dim3(256), 0, stream>>>(stats);
  fused_conv_mix_k<<<dim3(NB*49), dim3(256), 0, stream>>>(hk, x, wmix, bmix, y, stats);
  finalize_stats_k<<<dim3(1), dim3(128), 0, stream>>>(stats);
  long long total4 = (long long)NB * NC * HWSZ / 4;
  bn_gelu_k<<<dim3((unsigned)((total4 + 255) / 256)), dim3(256), 0, stream>>>(y, stats, total4);
}